// QPSolver_37452114821632
// MI455X (gfx1250) — compile-verified
//
#include <hip/hip_runtime.h>

// ---------------------------------------------------------------------------
// PDHG QP solver for MI455X (gfx1250, wave32, WMMA).
// Hot loop uses V_WMMA_F32_16X16X4_F32 with A (256x256) resident in LDS.
// ---------------------------------------------------------------------------

#define N_DIM 64
#define M_DIM 128
#define BSZ   256
#define ALPHA_C 1.0f
#define BETA_C  1.0f

typedef float v2f __attribute__((ext_vector_type(2)));
typedef float v8f __attribute__((ext_vector_type(8)));

union F8 { v8f v; float f[8]; };

#define ATP 264   // LDS pitch for A^T rows (264 % 64 == 8 -> half-waves hit disjoint banks)
#define XP  263   // LDS pitch for X rows  (263 % 64 == 7 -> conflict-free strided reads)

// ---------------------------------------------------------------------------
// Setup kernels (one-off, tiny cost)
// ---------------------------------------------------------------------------

// Gauss-Jordan inverse of an SPD SxS matrix (no pivoting needed for SPD).
// Single workgroup; augmented matrix in dynamic LDS: S x (2S+1) + S factors.
__global__ void spd_inverse(const float* __restrict__ Ain,
                            float* __restrict__ Out, int S) {
  extern __shared__ float aug[];
  const int P2 = 2 * S + 1;
  float* fac = aug + S * P2;
  const int tid = threadIdx.x, nt = blockDim.x;
  for (int idx = tid; idx < S * S; idx += nt) {
    int r = idx / S, c = idx % S;
    aug[r * P2 + c]     = Ain[idx];
    aug[r * P2 + S + c] = (r == c) ? 1.f : 0.f;
  }
  __syncthreads();
  for (int p = 0; p < S; ++p) {
    const float inv = 1.0f / aug[p * P2 + p];
    __syncthreads();
    for (int c = tid; c < 2 * S; c += nt) aug[p * P2 + c] *= inv;
    __syncthreads();
    for (int r = tid; r < S; r += nt) fac[r] = aug[r * P2 + p];
    __syncthreads();
    for (int idx = tid; idx < S * 2 * S; idx += nt) {
      int r = idx / (2 * S), c = idx % (2 * S);
      if (r != p) aug[r * P2 + c] -= fac[r] * aug[p * P2 + c];
    }
    __syncthreads();
  }
  for (int idx = tid; idx < S * S; idx += nt)
    Out[idx] = aug[(idx / S) * P2 + S + (idx % S)];
}

// C[MxN] = alpha * opA(A) * opB(B) + cscale*Cin + diag*(i==j)
__global__ void sgemm_naive(const float* __restrict__ A, const float* __restrict__ B,
                            const float* __restrict__ Cin, float* __restrict__ C,
                            int M, int N, int K, int tA, int tB,
                            float alpha, float cscale, float diag) {
  int idx = blockIdx.x * blockDim.x + threadIdx.x;
  if (idx >= M * N) return;
  int i = idx / N, j = idx % N;
  float s = 0.f;
  for (int k = 0; k < K; ++k) {
    float av = tA ? A[k * M + i] : A[i * K + k];
    float bv = tB ? B[j * K + k] : B[k * N + j];
    s = fmaf(av, bv, s);
  }
  float c = alpha * s + ((i == j) ? diag : 0.f);
  if (Cin) c += cscale * Cin[idx];
  C[idx] = c;
}

// out[c*R + r] = in[r*C + c]
__global__ void transpose_k(const float* __restrict__ in, float* __restrict__ out,
                            int R, int C) {
  int idx = blockIdx.x * blockDim.x + threadIdx.x;
  if (idx >= R * C) return;
  int r = idx / C, c = idx % C;
  out[c * R + r] = in[idx];
}

// Build A^T (256x256, At[k][j] = A[j][k]) from tD (128x128), using tDD = BETA*tD.
// A = [[tDD, tD], [I - 2a*tDD, I - 2a*tD]]
__global__ void build_At(const float* __restrict__ tD, float* __restrict__ At) {
  int idx = blockIdx.x * blockDim.x + threadIdx.x;
  if (idx >= 256 * 256) return;
  int k = idx >> 8, j = idx & 255;
  const float a2 = 2.f * ALPHA_C;
  float v;
  if (j < M_DIM) {
    v = (k < M_DIM) ? (BETA_C * tD[j * M_DIM + k]) : tD[j * M_DIM + (k - M_DIM)];
  } else {
    int jj = j - M_DIM;
    if (k < M_DIM) v = -a2 * BETA_C * tD[jj * M_DIM + k] + ((jj == k) ? 1.f : 0.f);
    else           v = ((jj == (k - M_DIM)) ? 1.f : 0.f) - a2 * tD[jj * M_DIM + (k - M_DIM)];
  }
  At[k * 256 + j] = v;
}

// ---------------------------------------------------------------------------
// Main PDHG iteration kernel.
// grid = 16 workgroups (one 16-row batch slab each), block = 512 threads (16 waves).
// Wave j computes the 16x16 output tile at columns [16j, 16j+16) each iteration
// via 64 chained v_wmma_f32_16x16x4_f32. A^T lives in LDS (264-float pitch);
// X double-buffers in LDS (263-float pitch). One barrier per iteration.
// Waves 0..3 additionally produce the primal recovery tiles (K=128, from L2-
// resident pinvH^T). All big outputs use non-temporal stores to keep the
// working set (A^T copies, mu, Pt) resident in the 192MB L2.
// ---------------------------------------------------------------------------
__global__ __launch_bounds__(512, 1)
void pdhg_main(const float* __restrict__ Atg,     // 256x256  A^T
               const float* __restrict__ mu,      // 256x128
               const float* __restrict__ Pt,      // 128x64   (pinvH)^T
               const float* __restrict__ pconst,  // 256x64   -(b @ pinvH^T)
               float* __restrict__ Xs,            // (256, iters+1, 256)
               float* __restrict__ Pr,            // (256, iters+1, 64)
               int iters) {
  extern __shared__ float smem[];
  float* sAt = smem;                // [256][ATP]
  float* sX  = smem + 256 * ATP;    // [2][16][XP]

  const int tid  = threadIdx.x;
  const int lane = tid & 31;
  const int wave = tid >> 5;        // 0..15
  const int b0   = blockIdx.x * 16; // batch slab base
  const int j0   = wave * 16;       // output column base

  // --- stage A^T into LDS (coalesced float4) ---
  for (int idx = tid; idx < 256 * 64; idx += 512) {
    int k  = idx >> 6;
    int j4 = (idx & 63) << 2;
    float4 v = reinterpret_cast<const float4*>(Atg)[idx];
    *reinterpret_cast<float4*>(&sAt[k * ATP + j4]) = v;
  }
  // --- zero X buffer 0 ---
  for (int idx = tid; idx < 16 * XP; idx += 512) sX[idx] = 0.f;

  // --- per-wave constant tiles (WMMA C/D layout: lane holds col c, rows r+8*half) ---
  const int cM = lane >> 4;   // row-half select
  const int cc = lane & 15;   // column within tile
  F8 bv;                      // Bvec tile: [mu | -2a*mu]
  for (int r = 0; r < 8; ++r) {
    int M = r + 8 * cM;
    bv.f[r] = (j0 < M_DIM)
        ? mu[(b0 + M) * M_DIM + j0 + cc]
        : -2.f * ALPHA_C * mu[(b0 + M) * M_DIM + (j0 - M_DIM) + cc];
  }
  F8 pc;
  if (wave < 4)
    for (int r = 0; r < 8; ++r)
      pc.f[r] = pconst[(b0 + r + 8 * cM) * N_DIM + j0 + cc];

  __syncthreads();

  const long long T1 = (long long)(iters + 1);

  // --- t = 0 outputs: X0 = 0, primal0 = pconst ---
  for (int r = 0; r < 8; ++r) {
    int M = r + 8 * cM;
    __builtin_nontemporal_store(0.f, Xs + ((long long)(b0 + M) * T1) * 256 + j0 + cc);
  }
  if (wave < 4)
    for (int r = 0; r < 8; ++r)
      __builtin_nontemporal_store(pc.f[r],
          Pr + ((long long)(b0 + r + 8 * cM) * T1) * N_DIM + j0 + cc);

  const int aM   = lane & 15;       // A-operand M row
  const int koff = 2 * (lane >> 4); // A/B-operand K offset for upper half-wave

  for (int t = 1; t <= iters; ++t) {
    const float* Xc = sX + ((t - 1) & 1) * 16 * XP;
    float*       Xn = sX + (t & 1) * 16 * XP;

    // Xn_tile = X_slab @ A^T_tile + Bvec_tile  (start accumulator at Bvec)
    F8 acc; acc.v = bv.v;
#pragma unroll 8
    for (int k0 = 0; k0 < 256; k0 += 4) {
      const int kb = k0 + koff;
      v2f a, bb;
      a.x  = Xc[aM * XP + kb];
      a.y  = Xc[aM * XP + kb + 1];
      bb.x = sAt[kb * ATP + j0 + cc];
      bb.y = sAt[(kb + 1) * ATP + j0 + cc];
      acc.v = __builtin_amdgcn_wmma_f32_16x16x4_f32(
          false, a, false, bb, (short)0, acc.v, false, false);
    }
    if (j0 >= M_DIM)
      for (int r = 0; r < 8; ++r) acc.f[r] = fmaxf(acc.f[r], 0.f);

    for (int r = 0; r < 8; ++r) {
      int M = r + 8 * cM;
      Xn[M * XP + j0 + cc] = acc.f[r];
      __builtin_nontemporal_store(acc.f[r],
          Xs + ((long long)(b0 + M) * T1 + t) * 256 + j0 + cc);
    }
    __syncthreads();

    // primal tile (waves 0..3): pa = Xn[:,128:] @ Pt + pconst
    if (wave < 4) {
      F8 pa; pa.v = pc.v;
#pragma unroll 8
      for (int k0 = 0; k0 < 128; k0 += 4) {
        const int kb = k0 + koff;
        v2f a, bb;
        a.x  = Xn[aM * XP + M_DIM + kb];
        a.y  = Xn[aM * XP + M_DIM + kb + 1];
        bb.x = Pt[kb * N_DIM + j0 + cc];
        bb.y = Pt[(kb + 1) * N_DIM + j0 + cc];
        pa.v = __builtin_amdgcn_wmma_f32_16x16x4_f32(
            false, a, false, bb, (short)0, pa.v, false, false);
      }
      for (int r = 0; r < 8; ++r)
        __builtin_nontemporal_store(pa.f[r],
            Pr + ((long long)(b0 + r + 8 * cM) * T1 + t) * N_DIM + j0 + cc);
    }
    // No second barrier needed: iter t+1 writes the *other* X buffer, and the
    // buffer read by primal-t is not rewritten until after barrier at t+1.
  }
}

// ---------------------------------------------------------------------------
extern "C" void kernel_launch(void* const* d_in, const int* in_sizes, int n_in,
                              void* d_out, int out_size, void* d_ws, size_t ws_size,
                              hipStream_t stream) {
  const float* q = (const float*)d_in[0];  // 256x64
  const float* b = (const float*)d_in[1];  // 256x128
  const float* P = (const float*)d_in[2];  // 64x64
  const float* H = (const float*)d_in[3];  // 128x64
  (void)n_in; (void)in_sizes; (void)ws_size;

  // out = Xs (256,(iters+1),256) ++ primal (256,(iters+1),64)
  int iters = out_size / (BSZ * (2 * M_DIM + N_DIM)) - 1;
  if (iters < 1) iters = 1000;

  float* w = (float*)d_ws;
  float* Pinv   = w;            // 4096
  float* PinvHt = w + 4096;     // 8192
  float* Mm     = w + 12288;    // 16384
  float* tD     = w + 28672;    // 16384
  float* Pinvq  = w + 45056;    // 16384
  float* t1     = w + 61440;    // 32768
  float* mu     = w + 94208;    // 32768
  float* Atg    = w + 126976;   // 65536
  float* G      = w + 192512;   // 4096
  float* Ginv   = w + 196608;   // 4096
  float* pinvH  = w + 200704;   // 8192
  float* Pt     = w + 208896;   // 8192
  float* pconst = w + 217088;   // 16384

  float* Xs = (float*)d_out;
  float* Pr = Xs + (size_t)(iters + 1) * BSZ * (2 * M_DIM);

  // ---- setup (one-off small linear algebra) ----
  spd_inverse<<<1, 256, (64 * 129 + 64) * 4, stream>>>(P, Pinv, 64);
  // PinvHt = Pinv @ H^T   (64x128)
  sgemm_naive<<<32, 256, 0, stream>>>(Pinv, H, nullptr, PinvHt, 64, 128, 64, 0, 1, 1.f, 0.f, 0.f);
  // Mm = BETA*I + H @ PinvHt  (128x128)
  sgemm_naive<<<64, 256, 0, stream>>>(H, PinvHt, nullptr, Mm, 128, 128, 64, 0, 0, 1.f, 0.f, BETA_C);
  spd_inverse<<<1, 256, (128 * 257 + 128) * 4, stream>>>(Mm, tD, 128);
  // Pinv_q = q @ Pinv (Pinv symmetric)  (256x64)
  sgemm_naive<<<64, 256, 0, stream>>>(q, Pinv, nullptr, Pinvq, 256, 64, 64, 0, 0, 1.f, 0.f, 0.f);
  // t1 = Pinv_q @ H^T - b   (256x128)
  sgemm_naive<<<128, 256, 0, stream>>>(Pinvq, H, b, t1, 256, 128, 64, 0, 1, 1.f, -1.f, 0.f);
  // mu = t1 @ tD^T          (256x128)
  sgemm_naive<<<128, 256, 0, stream>>>(t1, tD, nullptr, mu, 256, 128, 128, 0, 1, 1.f, 0.f, 0.f);
  build_At<<<256, 256, 0, stream>>>(tD, Atg);
  // pinv(H) for tall full-column-rank H: (H^T H)^{-1} H^T
  sgemm_naive<<<16, 256, 0, stream>>>(H, H, nullptr, G, 64, 64, 128, 1, 0, 1.f, 0.f, 0.f);
  spd_inverse<<<1, 256, (64 * 129 + 64) * 4, stream>>>(G, Ginv, 64);
  sgemm_naive<<<32, 256, 0, stream>>>(Ginv, H, nullptr, pinvH, 64, 128, 64, 0, 1, 1.f, 0.f, 0.f);
  transpose_k<<<32, 256, 0, stream>>>(pinvH, Pt, 64, 128);
  // pconst = -(b @ pinvH^T)  (256x64)
  sgemm_naive<<<64, 256, 0, stream>>>(b, pinvH, nullptr, pconst, 256, 64, 128, 0, 1, -1.f, 0.f, 0.f);

  // ---- main sequential recurrence: 16 independent batch slabs ----
  const size_t ldsBytes = (size_t)(256 * ATP + 2 * 16 * XP) * sizeof(float); // ~297 KB
  pdhg_main<<<16, 512, ldsBytes, stream>>>(Atg, mu, Pt, pconst, Xs, Pr, iters);
}